// Model_56977036149266
// MI455X (gfx1250) — compile-verified
//
#include <hip/hip_runtime.h>
#include <math.h>
#include <stdint.h>

typedef __attribute__((ext_vector_type(2))) float v2f;
typedef __attribute__((ext_vector_type(8))) float v8f;

#define VOCAB   100000
#define D       300
#define BATCH   4096
#define RQ      32      // queries per batch (N dim of GEMM)
#define SSUP    64      // 2 channels * 32 support vectors (M dim of GEMM)
#define LSTR    302     // LDS row stride in floats (even -> 8B aligned pairs)
#define EPS     1e-8f

// ---------------------------------------------------------------------------
// CDNA5 async global->LDS copy (ASYNCcnt-tracked), 8 bytes per lane.
// ---------------------------------------------------------------------------
__device__ inline void async_g2l_b64(uint32_t lds_byte_off, const void* gaddr) {
    asm volatile("global_load_async_to_lds_b64 %0, %1, off"
                 :: "v"(lds_byte_off), "v"((uint64_t)(uintptr_t)gaddr)
                 : "memory");
}
__device__ inline void wait_asynccnt0() {
    asm volatile("s_wait_asynccnt 0x0" ::: "memory");
}

// ---------------------------------------------------------------------------
// Kernel 1: normalize the 64 shared support vectors into workspace.
// ---------------------------------------------------------------------------
__global__ __launch_bounds__(32) void support_norm_kernel(
    const int* __restrict__ support_idx,   // [64]
    const float* __restrict__ emb,         // [V, 300]
    float* __restrict__ usv)               // [64, 300]
{
    const int s    = blockIdx.x;           // 0..63
    const int lane = threadIdx.x;          // 0..31
    const size_t base = (size_t)support_idx[s] * D;

    float p = 0.f;
    for (int k = lane; k < D; k += 32) {
        float v = emb[base + k];
        p += v * v;
    }
    for (int off = 16; off > 0; off >>= 1) p += __shfl_xor(p, off, 32);
    const float scale = 1.f / fmaxf(sqrtf(p), EPS);
    for (int k = lane; k < D; k += 32)
        usv[s * D + k] = emb[base + k] * scale;
}

// ---------------------------------------------------------------------------
// Block-wide reductions (256 threads = 8 waves, wave32)
// ---------------------------------------------------------------------------
__device__ inline float wave_sum(float v) {
    for (int off = 16; off > 0; off >>= 1) v += __shfl_xor(v, off, 32);
    return v;
}
__device__ inline float wave_max(float v) {
    for (int off = 16; off > 0; off >>= 1) v = fmaxf(v, __shfl_xor(v, off, 32));
    return v;
}
__device__ inline float block_sum(float v, volatile float* red, int tid) {
    __syncthreads();
    v = wave_sum(v);
    if ((tid & 31) == 0) red[tid >> 5] = v;
    __syncthreads();
    float r = 0.f;
    #pragma unroll
    for (int i = 0; i < 8; ++i) r += red[i];
    return r;
}
__device__ inline float block_max(float v, volatile float* red, int tid) {
    __syncthreads();
    v = wave_max(v);
    if ((tid & 31) == 0) red[tid >> 5] = v;
    __syncthreads();
    float r = red[0];
    #pragma unroll
    for (int i = 1; i < 8; ++i) r = fmaxf(r, red[i]);
    return r;
}

// ---------------------------------------------------------------------------
// Kernel 2: one workgroup per batch. Async-gather+normalize queries, f32 WMMA
// cosine-sim GEMM (64x32, K=300), then the fused head entirely in LDS.
// ---------------------------------------------------------------------------
__global__ __launch_bounds__(256) void fewshot_head_kernel(
    const int*   __restrict__ tokens,    // [B, 32]
    const float* __restrict__ emb,       // [V, 300]
    const float* __restrict__ usv_g,     // [64, 300] pre-normalized
    const float* __restrict__ w1,        // [2,2,3,3]
    const float* __restrict__ b1,        // [2]
    const float* __restrict__ w2,        // [2,2,2,2]
    const float* __restrict__ b2,        // [2]
    const float* __restrict__ fc2w,      // [2,2]
    const float* __restrict__ fc2b,      // [2]
    const float* __restrict__ fcw,       // [2,128]
    const float* __restrict__ fcb,       // [2]
    float* __restrict__ out)             // [B, 2]
{
    extern __shared__ float smem[];
    // LDS carve-up (floats)
    float* usv   = smem;                         // 64*302 = 19328
    float* xq    = usv   + SSUP * LSTR;          // 32*302 =  9664
    float* smm   = xq    + RQ   * LSTR;          // 64*33  =  2112
    float* p1pad = smm   + SSUP * 33;            // 2*18*18=   648
    float* wbuf  = p1pad + 2 * 18 * 18;          // 320
    int*   tokl  = (int*)(wbuf + 320);           // 32
    float* invn  = (float*)(tokl + 32);          // 32
    float* red   = invn + 32;                    // 8

    const int tid  = threadIdx.x;
    const int b    = blockIdx.x;
    const int lane = tid & 31;
    const int wave = tid >> 5;

    // ---- stage tokens first (needed for gather addresses) ------------------
    if (tid < 32) tokl[tid] = tokens[b * RQ + tid];
    __syncthreads();

    // ---- async global->LDS staging: support matrix + query gather ----------
    // support: 64 rows * 150 b64 chunks
    for (int i = tid; i < SSUP * (D / 2); i += 256) {
        const int s = i / (D / 2), c = i - s * (D / 2);
        async_g2l_b64((uint32_t)(uintptr_t)&usv[s * LSTR + c * 2],
                      usv_g + s * D + c * 2);
    }
    // queries: 32 rows * 150 b64 chunks, gathered via token ids
    for (int i = tid; i < RQ * (D / 2); i += 256) {
        const int r = i / (D / 2), c = i - r * (D / 2);
        async_g2l_b64((uint32_t)(uintptr_t)&xq[r * LSTR + c * 2],
                      emb + (size_t)tokl[r] * D + c * 2);
    }

    // ---- overlap: stage weights, zero padded conv2 input -------------------
    if (tid < 36) wbuf[tid]       = w1[tid];          // W1 @0
    if (tid < 2)  wbuf[36 + tid]  = b1[tid];          // B1 @36
    if (tid < 16) wbuf[38 + tid]  = w2[tid];          // W2 @38
    if (tid < 2)  wbuf[54 + tid]  = b2[tid];          // B2 @54
    if (tid < 4)  wbuf[56 + tid]  = fc2w[tid];        // FC2W @56
    if (tid < 2)  wbuf[60 + tid]  = fc2b[tid];        // FC2B @60
    if (tid < 2)  wbuf[318 + tid] = fcb[tid];         // FCB @318
    for (int i = tid; i < 256; i += 256) wbuf[62 + i] = fcw[i];   // FCW @62
    for (int i = tid; i < 2 * 18 * 18; i += 256) p1pad[i] = 0.f;

    wait_asynccnt0();          // drain this wave's async LDS writes
    __syncthreads();           // make them visible block-wide

    // ---- per-query L2 norms (8 lanes per row) ------------------------------
    {
        const int r = tid >> 3, j = tid & 7;
        float p = 0.f;
        for (int k = j; k < D; k += 8) {
            float v = xq[r * LSTR + k];
            p += v * v;
        }
        p += __shfl_xor(p, 4, 32);
        p += __shfl_xor(p, 2, 32);
        p += __shfl_xor(p, 1, 32);
        if (j == 0) invn[r] = 1.f / fmaxf(sqrtf(p), EPS);
    }
    __syncthreads();
    for (int i = tid; i < RQ * D; i += 256) {
        const int r = i / D, k = i - r * D;
        xq[r * LSTR + k] *= invn[r];
    }
    __syncthreads();

    // ---- 64x32 cosine-sim GEMM: 8 waves, one 16x16 tile each, K=300 --------
    {
        const int mt   = wave >> 1;          // 0..3 -> support row tile
        const int nt   = wave & 1;           // 0..1 -> query col tile
        const int half = lane >> 4;          // K-pair select
        const int lm   = lane & 15;
        const float* arow = &usv[(mt * 16 + lm) * LSTR + half * 2];
        const float* brow = &xq[(nt * 16 + lm) * LSTR + half * 2];
        v8f acc = {};
        #pragma unroll 5
        for (int k0 = 0; k0 < D; k0 += 4) {
            v2f a = *(const v2f*)(arow + k0);
            v2f bf = *(const v2f*)(brow + k0);
            acc = __builtin_amdgcn_wmma_f32_16x16x4_f32(
                false, a, false, bf, (short)0, acc, false, false);
        }
        #pragma unroll
        for (int v = 0; v < 8; ++v) {
            const int m = mt * 16 + v + half * 8;
            smm[m * 33 + nt * 16 + lm] = acc[v];
        }
    }
    __syncthreads();

    // ---- channel bi-attention: softmax-pool + sigmoid gate -----------------
    float gate[2];
    {
        float pooled[2];
        #pragma unroll
        for (int n = 0; n < 2; ++n) {
            float mx = -INFINITY;
            for (int i = tid; i < 1024; i += 256)
                mx = fmaxf(mx, smm[(n * 32 + (i >> 5)) * 33 + (i & 31)]);
            mx = block_max(mx, red, tid);
            float se = 0.f, sw = 0.f;
            for (int i = tid; i < 1024; i += 256) {
                float v = smm[(n * 32 + (i >> 5)) * 33 + (i & 31)];
                float e = expf(v - mx);
                se += e; sw += e * v;
            }
            se = block_sum(se, red, tid);
            sw = block_sum(sw, red, tid);
            pooled[n] = sw / se;
        }
        #pragma unroll
        for (int c = 0; c < 2; ++c) {
            float z = pooled[0] * wbuf[56 + c * 2 + 0]
                    + pooled[1] * wbuf[56 + c * 2 + 1] + wbuf[60 + c];
            gate[c] = 1.f / (1.f + expf(-z));
        }
    }
    __syncthreads();
    for (int i = tid; i < 2048; i += 256) {
        const int n = i >> 10, s = (i >> 5) & 31, r = i & 31;
        smm[(n * 32 + s) * 33 + r] *= gate[n];
    }
    __syncthreads();

    // ---- conv1 3x3 pad1 + relu + maxpool2x2 -> p1pad interior --------------
    for (int j = tid; j < 2 * 16 * 16; j += 256) {
        const int o = j >> 8, rem = j & 255, py = rem >> 4, px = rem & 15;
        float m = -INFINITY;
        #pragma unroll
        for (int dy = 0; dy < 2; ++dy)
        #pragma unroll
        for (int dx = 0; dx < 2; ++dx) {
            const int y = 2 * py + dy, x = 2 * px + dx;
            float acc = wbuf[36 + o];
            #pragma unroll
            for (int i = 0; i < 2; ++i)
            #pragma unroll
            for (int ky = 0; ky < 3; ++ky) {
                const int yy = y + ky - 1;
                if (yy < 0 || yy >= 32) continue;
                #pragma unroll
                for (int kx = 0; kx < 3; ++kx) {
                    const int xx = x + kx - 1;
                    if (xx < 0 || xx >= 32) continue;
                    acc += wbuf[o * 18 + i * 9 + ky * 3 + kx] *
                           smm[(i * 32 + yy) * 33 + xx];
                }
            }
            m = fmaxf(m, fmaxf(acc, 0.f));
        }
        p1pad[o * 324 + (1 + py) * 18 + (1 + px)] = m;
    }
    __syncthreads();

    // ---- conv2 2x2 pad1 + relu + avgpool2x2 + final FC ---------------------
    float part0 = 0.f, part1 = 0.f;
    if (tid < 128) {
        const int n = tid >> 6, rem = tid & 63, py = rem >> 3, px = rem & 7;
        float s = 0.f;
        #pragma unroll
        for (int dy = 0; dy < 2; ++dy)
        #pragma unroll
        for (int dx = 0; dx < 2; ++dx) {
            const int y = 2 * py + dy, x = 2 * px + dx;   // conv2 out coords
            float c2 = wbuf[54 + n];
            #pragma unroll
            for (int i = 0; i < 2; ++i)
            #pragma unroll
            for (int ky = 0; ky < 2; ++ky)
            #pragma unroll
            for (int kx = 0; kx < 2; ++kx)
                c2 += wbuf[38 + n * 8 + i * 4 + ky * 2 + kx] *
                      p1pad[i * 324 + (y + ky) * 18 + (x + kx)];
            s += fmaxf(c2, 0.f);
        }
        const float val = 0.25f * s;
        part0 = wbuf[62 + tid]       * val;
        part1 = wbuf[62 + 128 + tid] * val;
    }
    const float o0 = block_sum(part0, red, tid);
    const float o1 = block_sum(part1, red, tid);
    if (tid == 0) {
        out[b * 2 + 0] = o0 + wbuf[318];
        out[b * 2 + 1] = o1 + wbuf[319];
    }
}

// ---------------------------------------------------------------------------
extern "C" void kernel_launch(void* const* d_in, const int* in_sizes, int n_in,
                              void* d_out, int out_size, void* d_ws, size_t ws_size,
                              hipStream_t stream) {
    const int*   tokens      = (const int*)  d_in[0];
    const int*   support_idx = (const int*)  d_in[1];
    const float* emb         = (const float*)d_in[2];
    const float* w1          = (const float*)d_in[3];
    const float* b1          = (const float*)d_in[4];
    const float* w2          = (const float*)d_in[5];
    const float* b2          = (const float*)d_in[6];
    const float* fc2w        = (const float*)d_in[7];
    const float* fc2b        = (const float*)d_in[8];
    const float* fcw         = (const float*)d_in[9];
    const float* fcb         = (const float*)d_in[10];
    float* usv = (float*)d_ws;               // 64*300 floats
    float* out = (float*)d_out;

    support_norm_kernel<<<SSUP, 32, 0, stream>>>(support_idx, emb, usv);

    const size_t lds_bytes =
        (size_t)(SSUP * LSTR + RQ * LSTR + SSUP * 33 + 2 * 18 * 18 + 320
                 + 32 + 32 + 8) * sizeof(float);
    fewshot_head_kernel<<<BATCH, 256, lds_bytes, stream>>>(
        tokens, emb, usv, w1, b1, w2, b2, fc2w, fc2b, fcw, fcb, out);
}